// RelposeRefinerModule_32358283608459
// MI455X (gfx1250) — compile-verified
//
#include <hip/hip_runtime.h>
#include <stdint.h>

// CDNA5 / gfx1250: wave32, WMMA (not MFMA), async global->LDS staging.
typedef __attribute__((ext_vector_type(2))) float v2f;
typedef __attribute__((ext_vector_type(8))) float v8f;

#define NPTS      8192
#define NT        256     // 8 wave32 waves per block
#define NW        8
#define CHUNKS    32      // 32-point chunks per wave: 8*32*32 = 8192
#define LM_ITERS  20
#define CHIR_MIN  0.01f
#define W_MIN     0.001f
#define LAM0      1e-3

// E = skew(t) * R   (f64)
__device__ __forceinline__ void essential_from_Rt(const double* R, const double* t, double* E) {
  for (int j = 0; j < 3; ++j) {
    E[0*3+j] = -t[2]*R[1*3+j] + t[1]*R[2*3+j];
    E[1*3+j] =  t[2]*R[0*3+j] - t[0]*R[2*3+j];
    E[2*3+j] = -t[1]*R[0*3+j] + t[0]*R[1*3+j];
  }
}

__device__ __forceinline__ float sampson_r(const float* __restrict__ E, float4 pt) {
  float Ex0 = E[0]*pt.x + E[1]*pt.y + E[2];
  float Ex1 = E[3]*pt.x + E[4]*pt.y + E[5];
  float Ex2 = E[6]*pt.x + E[7]*pt.y + E[8];
  float Et0 = E[0]*pt.z + E[3]*pt.w + E[6];
  float Et1 = E[1]*pt.z + E[4]*pt.w + E[7];
  float num = pt.z*Ex0 + pt.w*Ex1 + Ex2;
  float den = Ex0*Ex0 + Ex1*Ex1 + Et0*Et0 + Et1*Et1 + 1e-18f;
  return num / sqrtf(den);
}

__global__ __launch_bounds__(NT) void relpose_lm_kernel(
    const float* __restrict__ points,   // [8192,4]  (u1,v1,u2,v2)
    const float* __restrict__ models,   // [64,3,4]  (R|t)
    const float* __restrict__ K1,       // [3,3]
    const float* __restrict__ K2,       // [3,3]
    const float* __restrict__ prob,     // [64,8192]
    const float* __restrict__ prob_pow, // [1]
    float* __restrict__ out)            // [64,3,4]
{
  const int mdl  = blockIdx.x;
  const int tid  = threadIdx.x;
  const int wid  = tid >> 5;       // wave id 0..7
  const int lane = tid & 31;

  // ---------------- LDS ----------------
  __shared__ float4 s_pts[NPTS];            // 128 KB, staged once via async DMA
  __shared__ float  s_slab[NW][32][16];     // per-wave WMMA operand staging (16 KB)
  __shared__ float  s_Gred[NW][64];         // per-wave 8x8 Gram partials
  __shared__ double s_G[64];                // reduced 8x8 Gram (f64)
  __shared__ double s_R[9], s_t[3], s_Rn[9], s_tn[3];
  __shared__ double s_lam, s_cost, s_cn;
  __shared__ float  s_E[9], s_Ei[6][9], s_En[9];
  __shared__ double s_cwave[NW];

  // ---- 1. async-stage points[8192][4] (128 KB) into LDS (ASYNCcnt path) ----
  {
    const unsigned long long gbase = (unsigned long long)(uintptr_t)points;
    #pragma unroll
    for (int i = 0; i < NPTS / NT; ++i) {
      int idx = tid + i * NT;
      uint32_t ldsoff = (uint32_t)(uintptr_t)(&s_pts[idx]);  // low 32 bits = LDS byte offset
      uint32_t voff   = (uint32_t)(idx * 16);
      asm volatile("global_load_async_to_lds_b128 %0, %1, %2 offset:0"
                   :: "v"(ldsoff), "v"(voff), "s"(gbase) : "memory");
    }
    // prefetch this model's prob row while the point DMA is in flight
    __builtin_prefetch(prob + mdl * NPTS + tid * 4, 0, 0);
    asm volatile("s_wait_asynccnt 0" ::: "memory");
  }

  // zero the WMMA slab once: cols 7..15 stay zero forever (pad columns of the Gram)
  for (int i = tid; i < NW * 32 * 16; i += NT) ((float*)s_slab)[i] = 0.0f;

  // uniform scalars
  const float f1 = 0.5f * (K1[0] + K1[4]);
  const float f2 = 0.5f * (K2[0] + K2[4]);
  const float s  = 0.5f * (1.0f / f1 + 1.0f / f2);
  const float s2 = s * s;
  const float pp = prob_pow[0];

  if (tid == 0) {
    for (int i = 0; i < 9; ++i) s_R[i] = (double)models[mdl*12 + (i/3)*4 + (i%3)];
    for (int i = 0; i < 3; ++i) s_t[i] = (double)models[mdl*12 + i*4 + 3];
    s_lam = LAM0;
  }
  __syncthreads();

  // ---- 2. chirality + robust weights (fixed for all LM iterations) ----
  float Rf[9], tf[3];
  for (int i = 0; i < 9; ++i) Rf[i] = (float)s_R[i];
  for (int i = 0; i < 3; ++i) tf[i] = (float)s_t[i];

  float wreg[CHUNKS];   // per-thread weights for its 32 points
  #pragma unroll 4
  for (int c = 0; c < CHUNKS; ++c) {
    int p = wid * 1024 + c * 32 + lane;
    float4 pt = s_pts[p];
    float a0 = Rf[0]*pt.x + Rf[1]*pt.y + Rf[2];
    float a1 = Rf[3]*pt.x + Rf[4]*pt.y + Rf[5];
    float a2 = Rf[6]*pt.x + Rf[7]*pt.y + Rf[8];
    float b0 = -pt.z, b1 = -pt.w, b2 = -1.0f;
    float aa = a0*a0 + a1*a1 + a2*a2;
    float bb = b0*b0 + b1*b1 + b2*b2;
    float ab = a0*b0 + a1*b1 + a2*b2;
    float r1 = -(a0*tf[0] + a1*tf[1] + a2*tf[2]);
    float r2 = -(b0*tf[0] + b1*tf[1] + b2*tf[2]);
    float det = aa*bb - ab*ab + 1e-18f;
    float z1 = (bb*r1 - ab*r2) / det;
    float z2 = (aa*r2 - ab*r1) / det;
    bool chir = (z1 > CHIR_MIN) && (z2 > CHIR_MIN);
    float pv = prob[mdl * NPTS + p];
    pv = pv > 0.0f ? pv : 0.0f;
    float wv = chir ? __powf(pv, pp) : 0.0f;
    wreg[c] = (wv < W_MIN) ? 0.0f : wv;
  }

  // ---- 3. initial cost ----
  if (tid == 0) {
    double E_[9];
    essential_from_Rt(s_R, s_t, E_);
    for (int i = 0; i < 9; ++i) s_E[i] = (float)E_[i];
  }
  __syncthreads();
  {
    double cs = 0.0;
    for (int c = 0; c < CHUNKS; ++c) {
      int p = wid * 1024 + c * 32 + lane;
      float r = sampson_r(s_E, s_pts[p]);
      cs += (double)(wreg[c] * s2 * log1pf(r * r / s2));
    }
    for (int off = 16; off > 0; off >>= 1) cs += __shfl_down(cs, off, 32);
    if (lane == 0) s_cwave[wid] = cs;
  }
  __syncthreads();
  if (tid == 0) {
    double tot = 0.0;
    for (int k = 0; k < NW; ++k) tot += s_cwave[k];
    s_cost = tot;
  }
  __syncthreads();

  // ---- 4. LM iterations ----
  for (int it = 0; it < LM_ITERS; ++it) {
    // 4a. one lane builds E and the six directional derivatives E'_i (f64 -> f32)
    if (tid == 0) {
      double E_[9];
      essential_from_Rt(s_R, s_t, E_);
      for (int i = 0; i < 9; ++i) s_E[i] = (float)E_[i];
      for (int i = 0; i < 3; ++i) {     // rotation dirs: E * skew(e_k) (column permute)
        s_Ei[0][i*3+0] = 0.0f;             s_Ei[0][i*3+1] =  (float)E_[i*3+2]; s_Ei[0][i*3+2] = -(float)E_[i*3+1];
        s_Ei[1][i*3+0] = -(float)E_[i*3+2]; s_Ei[1][i*3+1] = 0.0f;             s_Ei[1][i*3+2] =  (float)E_[i*3+0];
        s_Ei[2][i*3+0] =  (float)E_[i*3+1]; s_Ei[2][i*3+1] = -(float)E_[i*3+0]; s_Ei[2][i*3+2] = 0.0f;
      }
      for (int j = 0; j < 3; ++j) {     // translation dirs: skew(e_k) * R (row permute)
        s_Ei[3][0*3+j] = 0.0f;                s_Ei[3][1*3+j] = -(float)s_R[2*3+j]; s_Ei[3][2*3+j] =  (float)s_R[1*3+j];
        s_Ei[4][0*3+j] =  (float)s_R[2*3+j];  s_Ei[4][1*3+j] = 0.0f;               s_Ei[4][2*3+j] = -(float)s_R[0*3+j];
        s_Ei[5][0*3+j] = -(float)s_R[1*3+j];  s_Ei[5][1*3+j] =  (float)s_R[0*3+j]; s_Ei[5][2*3+j] = 0.0f;
      }
    }
    __syncthreads();

    // 4b. residual+Jacobian pass; Gram accumulated with v_wmma_f32_16x16x4_f32.
    // Two accumulators break the D->C RAW chain between back-to-back WMMAs.
    v8f acc0 = {0.f, 0.f, 0.f, 0.f, 0.f, 0.f, 0.f, 0.f};
    v8f acc1 = {0.f, 0.f, 0.f, 0.f, 0.f, 0.f, 0.f, 0.f};
    const int hi = lane >> 4, lo = lane & 15;
    for (int c = 0; c < CHUNKS; ++c) {
      int p = wid * 1024 + c * 32 + lane;
      float4 pt = s_pts[p];
      float Ex0 = s_E[0]*pt.x + s_E[1]*pt.y + s_E[2];
      float Ex1 = s_E[3]*pt.x + s_E[4]*pt.y + s_E[5];
      float Ex2 = s_E[6]*pt.x + s_E[7]*pt.y + s_E[8];
      float Et0 = s_E[0]*pt.z + s_E[3]*pt.w + s_E[6];
      float Et1 = s_E[1]*pt.z + s_E[4]*pt.w + s_E[7];
      float num = pt.z*Ex0 + pt.w*Ex1 + Ex2;
      float den = Ex0*Ex0 + Ex1*Ex1 + Et0*Et0 + Et1*Et1 + 1e-18f;
      float sden = sqrtf(den);
      float r = num / sden;
      float invsden = 1.0f / sden;
      float inv2den = 0.5f / den;
      float irls = wreg[c] / (1.0f + r * r / s2);
      float sc = sqrtf(irls);
      #pragma unroll
      for (int i = 0; i < 6; ++i) {     // J_i = d(sampson)/d delta_i at 0 (matches jacfwd)
        const float* Ei = s_Ei[i];
        float p0 = Ei[0]*pt.x + Ei[1]*pt.y + Ei[2];
        float p1 = Ei[3]*pt.x + Ei[4]*pt.y + Ei[5];
        float p2 = Ei[6]*pt.x + Ei[7]*pt.y + Ei[8];
        float q0 = Ei[0]*pt.z + Ei[3]*pt.w + Ei[6];
        float q1 = Ei[1]*pt.z + Ei[4]*pt.w + Ei[7];
        float dnum = pt.z*p0 + pt.w*p1 + p2;
        float dden = 2.0f * (Ex0*p0 + Ex1*p1 + Et0*q0 + Et1*q1);
        s_slab[wid][lane][i] = sc * ((dnum - num * dden * inv2den) * invsden);
      }
      s_slab[wid][lane][6] = sc * r;    // col 6 = residual -> Gram col 6 = J^T W r

      // 8 WMMAs consume the 32-point block, 4 points each. A = block^T, B = block:
      // A[m][k] and B[k][n] share the identical lane/VGPR placement, so one v2f feeds both.
      // Software pipeline: issue all 8 operand loads first, then the WMMA chain,
      // so the DS waits amortize instead of serializing load->wait->wmma.
      v2f ab[8];
      #pragma unroll
      for (int k = 0; k < 8; ++k) {
        int r0 = 4 * k + 2 * hi;
        ab[k].x = s_slab[wid][r0    ][lo];
        ab[k].y = s_slab[wid][r0 + 1][lo];
      }
      #pragma unroll
      for (int k = 0; k < 8; k += 2) {
        acc0 = __builtin_amdgcn_wmma_f32_16x16x4_f32(
            false, ab[k],     false, ab[k],     (short)0, acc0, false, false);
        acc1 = __builtin_amdgcn_wmma_f32_16x16x4_f32(
            false, ab[k + 1], false, ab[k + 1], (short)0, acc1, false, false);
      }
    }
    v8f acc = acc0 + acc1;

    // 4c. cross-wave Gram reduction (promote to f64)
    if (lane < 8) {
      #pragma unroll
      for (int mm = 0; mm < 8; ++mm) s_Gred[wid][mm * 8 + lane] = acc[mm];
    }
    __syncthreads();
    if (tid < 64) {
      double g = 0.0;
      for (int k = 0; k < NW; ++k) g += (double)s_Gred[k][tid];
      s_G[tid] = g;
    }
    __syncthreads();

    // 4d. one lane: (JTJ + lam*I) delta = -g ; pivoted Gaussian elimination in f64
    if (tid == 0) {
      double A[6][7];
      for (int i = 0; i < 6; ++i) {
        for (int j = 0; j < 6; ++j) A[i][j] = s_G[i * 8 + j];
        A[i][i] += s_lam;
        A[i][6] = s_G[i * 8 + 6];       // g_i
      }
      for (int col = 0; col < 6; ++col) {
        int piv = col; double mx = fabs(A[col][col]);
        for (int rr = col + 1; rr < 6; ++rr)
          if (fabs(A[rr][col]) > mx) { mx = fabs(A[rr][col]); piv = rr; }
        if (piv != col)
          for (int j = col; j < 7; ++j) { double tmp = A[col][j]; A[col][j] = A[piv][j]; A[piv][j] = tmp; }
        double inv = 1.0 / (A[col][col] + 1e-300);
        for (int rr = col + 1; rr < 6; ++rr) {
          double f = A[rr][col] * inv;
          for (int j = col; j < 7; ++j) A[rr][j] -= f * A[col][j];
        }
      }
      double x[6];
      for (int i = 5; i >= 0; --i) {
        double acc2 = A[i][6];
        for (int j = i + 1; j < 6; ++j) acc2 -= A[i][j] * x[j];
        x[i] = acc2 / (A[i][i] + 1e-300);
      }
      double dx = -x[0], dy = -x[1], dz = -x[2];        // delta = -solve(...)
      double th = sqrt(dx*dx + dy*dy + dz*dz + 1e-24);
      double kx = dx / th, ky = dy / th, kz = dz / th;
      double st = sin(th), ct = 1.0 - cos(th);
      double K[9]  = {0, -kz, ky,  kz, 0, -kx,  -ky, kx, 0};
      double KK[9];
      for (int i = 0; i < 3; ++i)
        for (int j = 0; j < 3; ++j) {
          double ssum = 0;
          for (int k = 0; k < 3; ++k) ssum += K[i*3+k] * K[k*3+j];
          KK[i*3+j] = ssum;
        }
      double Rr[9];
      for (int i = 0; i < 9; ++i) Rr[i] = st * K[i] + ct * KK[i];
      Rr[0] += 1.0; Rr[4] += 1.0; Rr[8] += 1.0;
      for (int i = 0; i < 3; ++i)
        for (int j = 0; j < 3; ++j) {
          double ssum = 0;
          for (int k = 0; k < 3; ++k) ssum += s_R[i*3+k] * Rr[k*3+j];
          s_Rn[i*3+j] = ssum;
        }
      for (int i = 0; i < 3; ++i) s_tn[i] = s_t[i] - x[3 + i];
      double En[9];
      essential_from_Rt(s_Rn, s_tn, En);
      for (int i = 0; i < 9; ++i) s_En[i] = (float)En[i];
    }
    __syncthreads();

    // 4e. cost with candidate pose
    {
      double cs = 0.0;
      for (int c = 0; c < CHUNKS; ++c) {
        int p = wid * 1024 + c * 32 + lane;
        float r = sampson_r(s_En, s_pts[p]);
        cs += (double)(wreg[c] * s2 * log1pf(r * r / s2));
      }
      for (int off = 16; off > 0; off >>= 1) cs += __shfl_down(cs, off, 32);
      if (lane == 0) s_cwave[wid] = cs;
    }
    __syncthreads();
    if (tid == 0) {
      double tot = 0.0;
      for (int k = 0; k < NW; ++k) tot += s_cwave[k];
      s_cn = tot;
      if (tot < s_cost) {
        for (int i = 0; i < 9; ++i) s_R[i] = s_Rn[i];
        for (int i = 0; i < 3; ++i) s_t[i] = s_tn[i];
        s_lam *= 0.1;
        s_cost = tot;
      } else {
        s_lam *= 10.0;
      }
    }
    __syncthreads();
  }

  // ---- 5. emit refined [R|t] as f32 ----
  if (tid < 12) {
    int row = tid >> 2, col = tid & 3;
    double v = (col < 3) ? s_R[row * 3 + col] : s_t[row];
    out[mdl * 12 + tid] = (float)v;
  }
}

extern "C" void kernel_launch(void* const* d_in, const int* in_sizes, int n_in,
                              void* d_out, int out_size, void* d_ws, size_t ws_size,
                              hipStream_t stream) {
  (void)in_sizes; (void)n_in; (void)out_size; (void)d_ws; (void)ws_size;
  const float* points   = (const float*)d_in[0];   // [8192,4]
  const float* models   = (const float*)d_in[1];   // [64,3,4]
  const float* K1       = (const float*)d_in[2];   // [3,3]
  const float* K2       = (const float*)d_in[3];   // [3,3]
  const float* prob     = (const float*)d_in[4];   // [64,8192]
  const float* prob_pow = (const float*)d_in[5];   // [1]
  // d_in[6] = loss_shape: unused by the reference forward math.
  relpose_lm_kernel<<<dim3(64), dim3(NT), 0, stream>>>(
      points, models, K1, K2, prob, prob_pow, (float*)d_out);
}